// DirectedMessagePassing_29411936043407
// MI455X (gfx1250) — compile-verified
//
#include <hip/hip_runtime.h>
#include <cstdint>
#include <cstddef>

// Problem sizes (static per the reference).
#define NN 40000      // N_NODES
#define NE 400000     // N_EDGES
#define NF 128        // NODE_F
#define EF 16         // EDGE_F
#define HID 64
#define STEPS 3

typedef float v2f __attribute__((ext_vector_type(2)));
typedef float v8f __attribute__((ext_vector_type(8)));
typedef unsigned int u32x4 __attribute__((ext_vector_type(4)));
typedef int i32x4 __attribute__((ext_vector_type(4)));
typedef int i32x8 __attribute__((ext_vector_type(8)));

// ---------------------------------------------------------------------------
// fp32 WMMA 16x16x4: D = A(16x4) * B(4x16) + C(16x16).  (confirmed lowering:
// v_wmma_f32_16x16x4_f32 in round-1 histogram.)  fp32 matrix math is the right
// precision here: workload is bandwidth-bound (~1.2 GB vs 32 GFLOP).
// ---------------------------------------------------------------------------
#if defined(__AMDGCN__) && __has_builtin(__builtin_amdgcn_wmma_f32_16x16x4_f32)
__device__ __forceinline__ v8f wmma4(v2f a, v2f b, v8f c) {
  return __builtin_amdgcn_wmma_f32_16x16x4_f32(false, a, false, b, (short)0, c,
                                               false, false);
}
#else
__device__ __forceinline__ v8f wmma4(v2f a, v2f b, v8f c) {
  c[0] += a[0] * b[0] + a[1] * b[1];  // placeholder; histogram reveals miss
  return c;
}
#endif

// ---------------------------------------------------------------------------
// Tensor Data Mover: DMA a linear run of n fp32 (n < 65536) from global into
// LDS.  D# built per CDNA5 ISA: group0 = {count=1,
// lds_addr, global_addr, type=2}; group1 = {data_size=4B, tensor_dim0=n,
// tensor_dim1=1, tile_dim0=n, stride0=n}; groups 2/3 zero (<=2-D tensor).
// Tracked by TENSORcnt; waited with s_wait_tensorcnt.
// ---------------------------------------------------------------------------
#if defined(__AMDGCN__) && __has_builtin(__builtin_amdgcn_tensor_load_to_lds) \
    && __has_builtin(__builtin_amdgcn_s_wait_tensorcnt)
#define HAVE_TDM 1
__device__ __forceinline__ void tdm_load_linear(unsigned lds_addr,
                                                const void* g, unsigned n) {
  unsigned long long ga = (unsigned long long)g;
  u32x4 g0;
  g0[0] = 1u;                                                // count=1 (user)
  g0[1] = lds_addr;                                          // LDS dest (bytes)
  g0[2] = (unsigned)ga;                                      // global_addr lo
  g0[3] = (unsigned)((ga >> 32) & 0x1FFFFFFu) | (2u << 30);  // addr hi | type=2
  i32x8 g1;
  g1[0] = (int)(2u << 16);                    // data_size code 2 = 4 bytes
  g1[1] = (int)((n & 0xFFFFu) << 16);         // tensor_dim0[15:0]
  g1[2] = (int)((n >> 16) | (1u << 16));      // tensor_dim0[31:16]|tensor_dim1=1
  g1[3] = (int)(n << 16);                     // tile_dim0 = n (fits 16 bits)
  g1[4] = 0;                                  // tile_dim1/2 unused (1-D tile)
  g1[5] = (int)n;                             // tensor_dim0_stride lo
  g1[6] = 0;
  g1[7] = 0;
  i32x4 gz = {0, 0, 0, 0};
#if defined(__clang_major__) && __clang_major__ >= 23
  i32x8 gz8 = {0, 0, 0, 0, 0, 0, 0, 0};      // 6-arg form (therock headers)
  __builtin_amdgcn_tensor_load_to_lds(g0, g1, gz, gz, gz8, 0);
#else
  __builtin_amdgcn_tensor_load_to_lds(g0, g1, gz, gz, 0);  // 5-arg (ROCm 7.2)
#endif
}
#endif

__device__ __forceinline__ float relu(float x) { return x > 0.f ? x : 0.f; }

// ---------------------------------------------------------------------------
// CSR construction (once per launch; reused across all steps)
// ---------------------------------------------------------------------------
__global__ __launch_bounds__(256) void k_degree(const int* __restrict__ src,
                                                const int* __restrict__ tgt,
                                                int* deg_src, int* deg_tgt) {
  int e = blockIdx.x * 256 + threadIdx.x;
  if (e < NE) {
    atomicAdd(&deg_src[src[e]], 1);
    atomicAdd(&deg_tgt[tgt[e]], 1);
  }
}

__global__ __launch_bounds__(1024) void k_exscan(const int* __restrict__ deg,
                                                 int* __restrict__ rp) {
  __shared__ int ssum[1024];
  const int tid = threadIdx.x;
  const int CH = (NN + 1023) / 1024;  // 40
  const int base = tid * CH;
  int s = 0;
  for (int i = 0; i < CH; i++) {
    int idx = base + i;
    s += (idx < NN) ? deg[idx] : 0;
  }
  ssum[tid] = s;
  __syncthreads();
  for (int off = 1; off < 1024; off <<= 1) {
    __syncthreads();
    int v = (tid >= off) ? ssum[tid - off] : 0;
    __syncthreads();
    ssum[tid] += v;
  }
  __syncthreads();
  int run = (tid == 0) ? 0 : ssum[tid - 1];
  for (int i = 0; i < CH; i++) {
    int idx = base + i;
    if (idx < NN) {
      rp[idx] = run;
      run += deg[idx];
    }
  }
  if (tid == 1023) rp[NN] = run;
}

__global__ __launch_bounds__(256) void k_scatter(const int* __restrict__ src,
                                                 const int* __restrict__ tgt,
                                                 const int* __restrict__ rp_s,
                                                 const int* __restrict__ rp_t,
                                                 int* cur_s, int* cur_t,
                                                 int* eid_s, int* eid_t) {
  int e = blockIdx.x * 256 + threadIdx.x;
  if (e < NE) {
    int s = src[e];
    eid_s[rp_s[s] + atomicAdd(&cur_s[s], 1)] = e;
    int t = tgt[e];
    eid_t[rp_t[t] + atomicAdd(&cur_t[t], 1)] = e;
  }
}

// Sort each adjacency list (avg len 10) -> deterministic fp reduction order.
__global__ __launch_bounds__(256) void k_sortlists(const int* __restrict__ rp,
                                                   int* eids) {
  int node = blockIdx.x * 256 + threadIdx.x;
  if (node >= NN) return;
  int b = rp[node], e = rp[node + 1];
  for (int i = b + 1; i < e; i++) {
    int v = eids[i];
    int j = i - 1;
    while (j >= b && eids[j] > v) {
      eids[j + 1] = eids[j];
      j--;
    }
    eids[j + 1] = v;
  }
}

// ---------------------------------------------------------------------------
// Segment sum by tgt via CSR gather: 64 threads/node, feature on lanes.
// ---------------------------------------------------------------------------
__global__ __launch_bounds__(256) void k_node_gather(
    const float* __restrict__ h, const int* __restrict__ rp,
    const int* __restrict__ eids, float* __restrict__ out) {
  int node = blockIdx.x * 4 + (threadIdx.x >> 6);
  int f = threadIdx.x & 63;
  float s = 0.f;
  int b = rp[node], e = rp[node + 1];
  for (int i = b; i < e; i++) s += h[eids[i] * HID + f];
  out[node * HID + f] = s;
}

// ---------------------------------------------------------------------------
// Edge init: h = relu(edge_attr @ W0 + b0). One wave per 16-edge tile.
// ---------------------------------------------------------------------------
__global__ __launch_bounds__(128) void k_edge_init(
    const float* __restrict__ ea, const float* __restrict__ W0,
    const float* __restrict__ b0, float* __restrict__ h) {
  const int lane = threadIdx.x & 31, w = threadIdx.x >> 5;
  const int half = lane >> 4, l16 = lane & 15;
  const int base = (blockIdx.x * 4 + w) * 16;  // 25000 tiles exactly
  for (int n0 = 0; n0 < HID; n0 += 16) {
    v8f acc = {0.f, 0.f, 0.f, 0.f, 0.f, 0.f, 0.f, 0.f};
    for (int k0 = 0; k0 < EF; k0 += 4) {
      int kk = k0 + 2 * half;
      v2f a, b;
      a[0] = ea[(base + l16) * EF + kk];
      a[1] = ea[(base + l16) * EF + kk + 1];
      b[0] = W0[kk * HID + n0 + l16];
      b[1] = W0[(kk + 1) * HID + n0 + l16];
      acc = wmma4(a, b, acc);
    }
    float bias = b0[n0 + l16];
    for (int v = 0; v < 8; v++) {
      h[(base + v + 8 * half) * HID + n0 + l16] = relu(acc[v] + bias);
    }
  }
}

// ---------------------------------------------------------------------------
// Fused edge update (one message-passing step), 2 waves/block, 16-edge tiles:
//   agg[e] = node_sum[src(e)] - sum(h over reverse-pair edges)   (exact)
//   h_out  = relu(relu([h|agg] @ W1 + b1) @ W2 + b2) + h
// W2 (16 KB) is DMA'd into LDS by the Tensor Data Mover (wave 0), overlapping
// phases A/B; layer-2 B fragments are then served from LDS.
// LDS: z 16.9K + t 8.7K + W2 16K = 42 KB.
// ---------------------------------------------------------------------------
__global__ __launch_bounds__(64) void k_edge_update(
    const float* __restrict__ h, const float* __restrict__ node_sum,
    const int* __restrict__ src, const int* __restrict__ tgt,
    const int* __restrict__ rp_src, const int* __restrict__ eid_src,
    const float* __restrict__ W1, const float* __restrict__ b1,
    const float* __restrict__ W2, const float* __restrict__ b2,
    float* __restrict__ h_out) {
  __shared__ float z[2][16][132];      // [h | agg], stride 132 (==4 mod 64)
  __shared__ float t[2][16][68];       // layer-1 activation
  __shared__ float w2lds[HID * HID];   // W2 staged via TDM

  const int lane = threadIdx.x & 31, w = threadIdx.x >> 5;
  const int half = lane >> 4, l16 = lane & 15;
  const int base = (blockIdx.x * 2 + w) * 16;  // 12500 blocks exactly

  __builtin_prefetch(W1, 0, 3);  // global_prefetch_b8 (W1 stays WGP$-resident)

#if defined(HAVE_TDM)
  if (w == 0) {  // wave-uniform branch: one TDM issue per block
    tdm_load_linear((unsigned)(uintptr_t)w2lds, W2, HID * HID);
  }
#else
  for (int i = threadIdx.x; i < HID * HID; i += 64) w2lds[i] = W2[i];
#endif

  // Phase A: build z = [h | node_sum[src] - rev_sum] in LDS.
  for (int r = 0; r < 16; r++) {
    const int e = base + r;
    const int s = src[e], tt = tgt[e];
    float a0 = node_sum[s * HID + lane];
    float a1 = node_sum[s * HID + 32 + lane];
    const int cb = rp_src[tt], ce = rp_src[tt + 1];
    for (int c = cb; c < ce; c++) {  // out-edges of tgt(e); avg 10 candidates
      int ed = eid_src[c];
      if (tgt[ed] == s) {  // exact reverse pair (handles multiplicity)
        a0 -= h[ed * HID + lane];
        a1 -= h[ed * HID + 32 + lane];
      }
    }
    z[w][r][HID + lane] = a0;
    z[w][r][HID + 32 + lane] = a1;
    z[w][r][lane] = h[e * HID + lane];
    z[w][r][32 + lane] = h[e * HID + 32 + lane];
  }
  __syncthreads();

  // Phase B: t = relu(z @ W1 + b1), K = 128.  (B fragments from global; W1
  // is 32 KB and read-shared by every block -> WGP$/L2 resident.)
  for (int n0 = 0; n0 < HID; n0 += 16) {
    v8f acc = {0.f, 0.f, 0.f, 0.f, 0.f, 0.f, 0.f, 0.f};
    for (int k0 = 0; k0 < 2 * HID; k0 += 4) {
      int kk = k0 + 2 * half;
      v2f a, b;
      a[0] = z[w][l16][kk];
      a[1] = z[w][l16][kk + 1];
      b[0] = W1[kk * HID + n0 + l16];
      b[1] = W1[(kk + 1) * HID + n0 + l16];
      acc = wmma4(a, b, acc);
    }
    float bias = b1[n0 + l16];
    for (int v = 0; v < 8; v++) t[w][v + 8 * half][n0 + l16] = relu(acc[v] + bias);
  }
#if defined(HAVE_TDM)
  if (w == 0) __builtin_amdgcn_s_wait_tensorcnt(0);  // W2 DMA complete
#endif
  __syncthreads();

  // Phase C: h_out = relu(t @ W2 + b2) + h  (W2 from LDS; residual from z).
  for (int n0 = 0; n0 < HID; n0 += 16) {
    v8f acc = {0.f, 0.f, 0.f, 0.f, 0.f, 0.f, 0.f, 0.f};
    for (int k0 = 0; k0 < HID; k0 += 4) {
      int kk = k0 + 2 * half;
      v2f a, b;
      a[0] = t[w][l16][kk];
      a[1] = t[w][l16][kk + 1];
      b[0] = w2lds[kk * HID + n0 + l16];
      b[1] = w2lds[(kk + 1) * HID + n0 + l16];
      acc = wmma4(a, b, acc);
    }
    float bias = b2[n0 + l16];
    for (int v = 0; v < 8; v++) {
      int r = v + 8 * half;
      h_out[(base + r) * HID + n0 + l16] =
          relu(acc[v] + bias) + z[w][r][n0 + l16];
    }
  }
}

// ---------------------------------------------------------------------------
// Fused readout: out = relu([x|atom_msg] @ W3 + b3) @ W4 + b4 (no final relu).
// 2 waves/block; W4 staged via TDM.  LDS: 25.1K + 8.7K + 16K = 50 KB.
// ---------------------------------------------------------------------------
__global__ __launch_bounds__(64) void k_readout(
    const float* __restrict__ x, const float* __restrict__ am,
    const float* __restrict__ W3, const float* __restrict__ b3,
    const float* __restrict__ W4, const float* __restrict__ b4,
    float* __restrict__ out) {
  __shared__ float zr[2][16][196];     // [x | atom_msg], K=192, stride 196
  __shared__ float tr[2][16][68];
  __shared__ float w4lds[HID * HID];

  const int lane = threadIdx.x & 31, w = threadIdx.x >> 5;
  const int half = lane >> 4, l16 = lane & 15;
  const int base = (blockIdx.x * 2 + w) * 16;  // 2500 tiles exactly

  __builtin_prefetch(W3, 0, 3);

#if defined(HAVE_TDM)
  if (w == 0) {
    tdm_load_linear((unsigned)(uintptr_t)w4lds, W4, HID * HID);
  }
#else
  for (int i = threadIdx.x; i < HID * HID; i += 64) w4lds[i] = W4[i];
#endif

  for (int r = 0; r < 16; r++) {
    const int node = base + r;
    for (int j = 0; j < 4; j++)
      zr[w][r][lane + 32 * j] = x[node * NF + lane + 32 * j];
    zr[w][r][NF + lane] = am[node * HID + lane];
    zr[w][r][NF + 32 + lane] = am[node * HID + 32 + lane];
  }
  __syncthreads();

  for (int n0 = 0; n0 < HID; n0 += 16) {  // layer 1, K = 192
    v8f acc = {0.f, 0.f, 0.f, 0.f, 0.f, 0.f, 0.f, 0.f};
    for (int k0 = 0; k0 < NF + HID; k0 += 4) {
      int kk = k0 + 2 * half;
      v2f a, b;
      a[0] = zr[w][l16][kk];
      a[1] = zr[w][l16][kk + 1];
      b[0] = W3[kk * HID + n0 + l16];
      b[1] = W3[(kk + 1) * HID + n0 + l16];
      acc = wmma4(a, b, acc);
    }
    float bias = b3[n0 + l16];
    for (int v = 0; v < 8; v++) tr[w][v + 8 * half][n0 + l16] = relu(acc[v] + bias);
  }
#if defined(HAVE_TDM)
  if (w == 0) __builtin_amdgcn_s_wait_tensorcnt(0);
#endif
  __syncthreads();

  for (int n0 = 0; n0 < HID; n0 += 16) {  // layer 2, K = 64, no relu
    v8f acc = {0.f, 0.f, 0.f, 0.f, 0.f, 0.f, 0.f, 0.f};
    for (int k0 = 0; k0 < HID; k0 += 4) {
      int kk = k0 + 2 * half;
      v2f a, b;
      a[0] = tr[w][l16][kk];
      a[1] = tr[w][l16][kk + 1];
      b[0] = w4lds[kk * HID + n0 + l16];
      b[1] = w4lds[(kk + 1) * HID + n0 + l16];
      acc = wmma4(a, b, acc);
    }
    float bias = b4[n0 + l16];
    for (int v = 0; v < 8; v++)
      out[(base + v + 8 * half) * HID + n0 + l16] = acc[v] + bias;
  }
}

// ---------------------------------------------------------------------------
// Host orchestration.  Workspace ~230 MB.  All launches on `stream`;
// graph-capture safe (only kernels + hipMemsetAsync).
// ---------------------------------------------------------------------------
extern "C" void kernel_launch(void* const* d_in, const int* in_sizes, int n_in,
                              void* d_out, int out_size, void* d_ws,
                              size_t ws_size, hipStream_t stream) {
  const float* x = (const float*)d_in[0];
  const int* ei = (const int*)d_in[1];  // [2, NE]
  const float* ea = (const float*)d_in[2];
  // d_in[3] = batch (unused by the reference computation)
  const float* W0 = (const float*)d_in[4];
  const float* b0 = (const float*)d_in[5];
  const float* W1 = (const float*)d_in[6];
  const float* b1 = (const float*)d_in[7];
  const float* W2 = (const float*)d_in[8];
  const float* b2 = (const float*)d_in[9];
  const float* W3 = (const float*)d_in[10];
  const float* b3 = (const float*)d_in[11];
  const float* W4 = (const float*)d_in[12];
  const float* b4 = (const float*)d_in[13];
  float* out = (float*)d_out;

  const int* src = ei;
  const int* tgt = ei + NE;

  size_t off = 0;
  auto carve = [&](size_t bytes) -> void* {
    void* q = (char*)d_ws + off;
    off += (bytes + 255) & ~(size_t)255;
    return q;
  };
  float* h_a = (float*)carve(sizeof(float) * (size_t)NE * HID);
  float* h_b = (float*)carve(sizeof(float) * (size_t)NE * HID);
  float* node_sum = (float*)carve(sizeof(float) * (size_t)NN * HID);
  float* atom_msg = (float*)carve(sizeof(float) * (size_t)NN * HID);
  int* rp_src = (int*)carve(sizeof(int) * (NN + 1));
  int* rp_tgt = (int*)carve(sizeof(int) * (NN + 1));
  int* eid_src = (int*)carve(sizeof(int) * NE);
  int* eid_tgt = (int*)carve(sizeof(int) * NE);
  int* counters = (int*)carve(sizeof(int) * 4 * NN);
  int* deg_src = counters;
  int* deg_tgt = counters + NN;
  int* cur_src = counters + 2 * NN;
  int* cur_tgt = counters + 3 * NN;

  // --- CSR build (rebuilt each launch; sorted lists => deterministic) ---
  hipMemsetAsync(counters, 0, sizeof(int) * 4 * NN, stream);
  const int EB = (NE + 255) / 256;
  k_degree<<<EB, 256, 0, stream>>>(src, tgt, deg_src, deg_tgt);
  k_exscan<<<1, 1024, 0, stream>>>(deg_src, rp_src);
  k_exscan<<<1, 1024, 0, stream>>>(deg_tgt, rp_tgt);
  k_scatter<<<EB, 256, 0, stream>>>(src, tgt, rp_src, rp_tgt, cur_src, cur_tgt,
                                    eid_src, eid_tgt);
  const int NB = (NN + 255) / 256;
  k_sortlists<<<NB, 256, 0, stream>>>(rp_src, eid_src);
  k_sortlists<<<NB, 256, 0, stream>>>(rp_tgt, eid_tgt);

  // --- h = relu(edge_attr @ W0 + b0) ---
  k_edge_init<<<NE / 64, 128, 0, stream>>>(ea, W0, b0, h_a);

  // --- 3 message-passing steps (double-buffered) ---
  float* hc = h_a;
  float* hn = h_b;
  for (int s = 0; s < STEPS; s++) {
    k_node_gather<<<NN / 4, 256, 0, stream>>>(hc, rp_tgt, eid_tgt, node_sum);
    k_edge_update<<<NE / 32, 64, 0, stream>>>(hc, node_sum, src, tgt, rp_src,
                                              eid_src, W1, b1, W2, b2, hn);
    float* tmp = hc;
    hc = hn;
    hn = tmp;
  }

  // --- readout ---
  k_node_gather<<<NN / 4, 256, 0, stream>>>(hc, rp_tgt, eid_tgt, atom_msg);
  k_readout<<<NN / 32, 64, 0, stream>>>(x, atom_msg, W3, b3, W4, b4, out);
}